// GVPConvLayer_81819126988958
// MI455X (gfx1250) — compile-verified
//
#include <hip/hip_runtime.h>
#include <math.h>

// ---------------------------------------------------------------------------
// CDNA5 (gfx1250) GVPConvLayer. fp32 WMMA (V_WMMA_F32_16X16X4_F32) for all
// scalar-channel GEMMs, edges/nodes tiled 16 per wave along M.
// K-loops are split per A-segment so every A fetch is a single-address-space,
// branch-free global_load_b64 or ds_load_b64 (no flat loads, no branch ladder).
// ---------------------------------------------------------------------------

typedef __attribute__((ext_vector_type(2))) float v2f;
typedef __attribute__((ext_vector_type(8))) float v8f;

__device__ __forceinline__ v2f ld2(const float* p) { return *(const v2f*)p; }
__device__ __forceinline__ float sigf(float x) { return 1.0f / (1.0f + __expf(-x)); }
__device__ __forceinline__ v8f zero8() { v8f z = {0.f,0.f,0.f,0.f,0.f,0.f,0.f,0.f}; return z; }

// D[16 x 16*NT] += A[16 x kbeg..kend] * B[.. x N] (B pre-packed k-pair layout:
// Bp[(k/2)*2N + n*2 + (k&1)]).  A via lambda af(k) -> {A[m][k],A[m][k+1]},
// m = lane&15, k = k0 + 2*(lane>>4).  kbeg/kend multiples of 4.
template<int NT, typename AF>
__device__ __forceinline__ void wmma_gemm(const AF& af, const float* __restrict__ Bp,
                                          int kbeg, int kend, int N, int ntile0,
                                          v8f* acc) {
  const int lane = threadIdx.x & 31;
  const int kb   = (lane >> 4) << 1;
  const int nc2  = ((ntile0 << 4) + (lane & 15)) * 2;
  for (int k0 = kbeg; k0 < kend; k0 += 4) {
    const int k = k0 + kb;
    v2f a = af(k);
    const float* bb = Bp + (size_t)(k >> 1) * (size_t)(2 * N) + nc2;
#pragma unroll
    for (int t = 0; t < NT; ++t) {
      v2f b = *(const v2f*)(bb + 32 * t);
      acc[t] = __builtin_amdgcn_wmma_f32_16x16x4_f32(false, a, false, b,
                                                     (short)0, acc[t], false, false);
    }
  }
}

// --------------------------- prep kernels ----------------------------------
__global__ void zero_kernel(float* p, long n) {
  long i = (long)blockIdx.x * blockDim.x + threadIdx.x;
  if (i < n) p[i] = 0.0f;
}

// pack W[K x N] row-major -> Bp[(k/2)*2N + n*2 + (k&1)], zero-pad k in [K,Kp4)
__global__ void pack_kernel(const float* __restrict__ W, float* __restrict__ Bp,
                            int K, int Kp4, int N) {
  int idx = blockIdx.x * 256 + threadIdx.x;
  int total = Kp4 * N;
  if (idx >= total) return;
  int k = idx / N, n = idx - k * N;
  float v = (k < K) ? W[(size_t)k * N + n] : 0.0f;
  Bp[(size_t)(k >> 1) * (size_t)(2 * N) + n * 2 + (k & 1)] = v;
}

// --------------------------- edge (message) kernel -------------------------
// 2 waves / block, 32 edges / block (16 per wave along WMMA M).
#define E_EPB 32
#define E_TPB 64

__global__ void __launch_bounds__(64) gvp_edge_kernel(
    const float* __restrict__ s_in, const float* __restrict__ v_in,
    const int* __restrict__ eidx,
    const float* __restrict__ e_s, const float* __restrict__ e_v,
    const float* __restrict__ wh0, const float* __restrict__ wsb0,
    const float* __restrict__ wv0, const float* __restrict__ wvb0,
    const float* __restrict__ wh1, const float* __restrict__ wsb1,
    const float* __restrict__ wv1, const float* __restrict__ wvb1,
    const float* __restrict__ wh2, const float* __restrict__ wsb2,
    const float* __restrict__ wv2, const float* __restrict__ wvb2,
    const float* __restrict__ Pws0, const float* __restrict__ Pws1,
    const float* __restrict__ Pws2, const float* __restrict__ Psv0,
    const float* __restrict__ Psv1, const float* __restrict__ Psv2,
    float* __restrict__ agg_s, float* __restrict__ agg_v,
    float* __restrict__ cnt, int nE) {
  __shared__ __align__(16) char smem[45696];
  int*   Lsrc = (int*)(smem);              // [32]
  int*   Ldst = (int*)(smem + 128);        // [32]
  float* SBUF = (float*)(smem + 256);      // [32][128] pre-activations; MV alias
  float* MV   = SBUF;                      // [32][99]
  float* VH   = (float*)(smem + 16640);    // [32][99] vh0; later vh2 [32][48]
  float* VHB  = (float*)(smem + 29312);    // [32][48] vh1; VN0 alias [32][36]
  float* VN0  = VHB;
  float* VB   = (float*)(smem + 35456);    // [32][48] v1/v2/v3
  float* VN1  = (float*)(smem + 41600);    // [32][16] vn1/vn2
  float* GB   = (float*)(smem + 43648);    // [32][16] sigmoid(gate)

  const int tid = threadIdx.x, lane = tid & 31, wid = tid >> 5;
  const int m16 = lane & 15, hi = lane >> 4;
  const int eblk = blockIdx.x * E_EPB;
  const int mW = wid * 16 + m16;
  const int e  = min(eblk + mW, nE - 1);
  const int srcI = eidx[e];
  const int dstI = eidx[nE + e];
  if (hi == 0) { Lsrc[mW] = srcI; Ldst[mW] = dstI; }
  __syncthreads();

  // ---- stage mv = [v_src(16) | edge_v(1) | v_dst(16)] x 3ch ----
  for (int t = tid; t < E_EPB * 99; t += E_TPB) {
    int m = t / 99, j = t - m * 99, vi = j / 3, c = j - vi * 3;
    int ee = min(eblk + m, nE - 1);
    float x;
    if (vi < 16)       x = v_in[(size_t)Lsrc[m] * 48 + vi * 3 + c];
    else if (vi == 16) x = e_v[(size_t)ee * 3 + c];
    else               x = v_in[(size_t)Ldst[m] * 48 + (vi - 17) * 3 + c];
    MV[t] = x;
  }
  __syncthreads();
  // ---- vh0 = mv @ wh0   [33,3] ----
  for (int t = tid; t < E_EPB * 99; t += E_TPB) {
    int m = t / 99, j = t - m * 99, h = j / 3, c = j - h * 3;
    float a = 0.f;
    for (int q = 0; q < 33; ++q) a = fmaf(MV[m * 99 + q * 3 + c], wh0[q * 33 + h], a);
    VH[t] = a;
  }
  __syncthreads();
  // ---- vn0 (padded to 36 with zeros for K-tail) ----
  for (int t = tid; t < E_EPB * 36; t += E_TPB) {
    int m = t / 36, h = t - m * 36;
    float x = 0.f;
    if (h < 33) {
      float a = VH[m*99 + h*3], b = VH[m*99 + h*3 + 1], c = VH[m*99 + h*3 + 2];
      x = sqrtf(fmaxf(a*a + b*b + c*c, 1e-8f));
    }
    VN0[m * 36 + h] = x;
  }
  __syncthreads();

  // ---- GVP0 scalar GEMM: [s_src|e_s|s_dst|vn0](321->324) @ ws0 -> 128 ----
  // Branch-free per-segment K loops (pure global or pure LDS A streams).
  v8f acc[8];
#pragma unroll
  for (int t = 0; t < 8; ++t) acc[t] = zero8();
  {
    const float* aS = s_in + (size_t)srcI * 128;
    const float* aE = e_s + (size_t)e * 32;
    const float* aD = s_in + (size_t)dstI * 128;
    wmma_gemm<8>([&](int k) -> v2f { return ld2(aS + k); },
                 Pws0, 0, 128, 128, 0, acc);
    wmma_gemm<8>([&](int k) -> v2f { return ld2(aE + (k - 128)); },
                 Pws0, 128, 160, 128, 0, acc);
    wmma_gemm<8>([&](int k) -> v2f { return ld2(aD + (k - 160)); },
                 Pws0, 160, 288, 128, 0, acc);
    wmma_gemm<8>([&](int k) -> v2f { return ld2(VN0 + mW * 36 + (k - 288)); },
                 Pws0, 288, 324, 128, 0, acc);
  }
#pragma unroll
  for (int t = 0; t < 8; ++t) {
    int col = t * 16 + m16; float bb = wsb0[col];
#pragma unroll
    for (int i = 0; i < 8; ++i) SBUF[(wid*16 + i + 8*hi) * 128 + col] = acc[t][i] + bb;
  }
  __syncthreads();

  // ---- gate0 = sigmoid(pre0) @ wsv0 ----
  v8f g = zero8();
  auto afsig = [&](int k) -> v2f {
    v2f x = ld2(SBUF + mW * 128 + k);
    v2f r; r[0] = sigf(x[0]); r[1] = sigf(x[1]); return r;
  };
  wmma_gemm<1>(afsig, Psv0, 0, 128, 16, 0, &g);
  {
    float bb = wvb0[m16];
#pragma unroll
    for (int i = 0; i < 8; ++i) GB[(wid*16 + i + 8*hi) * 16 + m16] = sigf(g[i] + bb);
  }
  __syncthreads();
  // ---- v1 = (vh0 @ wv0) * sigmoid(gate0) ----
  for (int t = tid; t < E_EPB * 48; t += E_TPB) {
    int m = t / 48, j = t - m * 48, o = j / 3, c = j - o * 3;
    float a = 0.f;
    for (int h = 0; h < 33; ++h) a = fmaf(VH[m*99 + h*3 + c], wv0[h*16 + o], a);
    VB[t] = a * GB[m * 16 + o];
  }
  __syncthreads();
  // ---- GVP1: vh1 = v1 @ wh1; vn1 ----
  for (int t = tid; t < E_EPB * 48; t += E_TPB) {
    int m = t / 48, j = t - m * 48, h = j / 3, c = j - h * 3;
    float a = 0.f;
    for (int o = 0; o < 16; ++o) a = fmaf(VB[m*48 + o*3 + c], wh1[o*16 + h], a);
    VHB[t] = a;
  }
  __syncthreads();
  for (int t = tid; t < E_EPB * 16; t += E_TPB) {
    int m = t >> 4, h = t & 15;
    float a = VHB[m*48 + h*3], b = VHB[m*48 + h*3 + 1], c = VHB[m*48 + h*3 + 2];
    VN1[t] = sqrtf(fmaxf(a*a + b*b + c*c, 1e-8f));
  }
  __syncthreads();
  // ---- GVP1 scalar GEMM: [relu(pre0)|vn1](144) @ ws1 ----
  auto afrel = [&](int k) -> v2f {
    v2f x = ld2(SBUF + mW * 128 + k);
    v2f r; r[0] = fmaxf(x[0], 0.f); r[1] = fmaxf(x[1], 0.f); return r;
  };
  auto afvn1 = [&](int k) -> v2f { return ld2(VN1 + mW * 16 + (k - 128)); };
#pragma unroll
  for (int t = 0; t < 8; ++t) acc[t] = zero8();
  wmma_gemm<8>(afrel, Pws1, 0, 128, 128, 0, acc);
  wmma_gemm<8>(afvn1, Pws1, 128, 144, 128, 0, acc);
  __syncthreads();
#pragma unroll
  for (int t = 0; t < 8; ++t) {
    int col = t * 16 + m16; float bb = wsb1[col];
#pragma unroll
    for (int i = 0; i < 8; ++i) SBUF[(wid*16 + i + 8*hi) * 128 + col] = acc[t][i] + bb;
  }
  __syncthreads();
  // ---- gate1 ----
  g = zero8();
  wmma_gemm<1>(afsig, Psv1, 0, 128, 16, 0, &g);
  {
    float bb = wvb1[m16];
#pragma unroll
    for (int i = 0; i < 8; ++i) GB[(wid*16 + i + 8*hi) * 16 + m16] = sigf(g[i] + bb);
  }
  __syncthreads();
  // ---- v2 = (vh1 @ wv1) * gate1 ----
  for (int t = tid; t < E_EPB * 48; t += E_TPB) {
    int m = t / 48, j = t - m * 48, o = j / 3, c = j - o * 3;
    float a = 0.f;
    for (int h = 0; h < 16; ++h) a = fmaf(VHB[m*48 + h*3 + c], wv1[h*16 + o], a);
    VB[t] = a * GB[m * 16 + o];
  }
  __syncthreads();
  // ---- GVP2: vh2 (-> VH region), vn2 (-> VN1 region) ----
  for (int t = tid; t < E_EPB * 48; t += E_TPB) {
    int m = t / 48, j = t - m * 48, h = j / 3, c = j - h * 3;
    float a = 0.f;
    for (int o = 0; o < 16; ++o) a = fmaf(VB[m*48 + o*3 + c], wh2[o*16 + h], a);
    VH[m * 99 + j] = a;
  }
  __syncthreads();
  for (int t = tid; t < E_EPB * 16; t += E_TPB) {
    int m = t >> 4, h = t & 15;
    float a = VH[m*99 + h*3], b = VH[m*99 + h*3 + 1], c = VH[m*99 + h*3 + 2];
    VN1[t] = sqrtf(fmaxf(a*a + b*b + c*c, 1e-8f));
  }
  __syncthreads();
  // ---- GVP2 scalar GEMM (acts=False): keep messages in acc ----
#pragma unroll
  for (int t = 0; t < 8; ++t) acc[t] = zero8();
  wmma_gemm<8>(afrel, Pws2, 0, 128, 128, 0, acc);
  wmma_gemm<8>(afvn1, Pws2, 128, 144, 128, 0, acc);
#pragma unroll
  for (int t = 0; t < 8; ++t) {
    int col = t * 16 + m16; float bb = wsb2[col];
#pragma unroll
    for (int i = 0; i < 8; ++i) acc[t][i] += bb;
  }
  __syncthreads();
#pragma unroll
  for (int t = 0; t < 8; ++t) {
    int col = t * 16 + m16;
#pragma unroll
    for (int i = 0; i < 8; ++i) SBUF[(wid*16 + i + 8*hi) * 128 + col] = acc[t][i];
  }
  __syncthreads();
  // ---- gate2: identity gate_in ----
  g = zero8();
  auto afid = [&](int k) -> v2f { return ld2(SBUF + mW * 128 + k); };
  wmma_gemm<1>(afid, Psv2, 0, 128, 16, 0, &g);
  {
    float bb = wvb2[m16];
#pragma unroll
    for (int i = 0; i < 8; ++i) GB[(wid*16 + i + 8*hi) * 16 + m16] = sigf(g[i] + bb);
  }
  __syncthreads();
  // ---- v3 = (vh2 @ wv2) * gate2 ----
  for (int t = tid; t < E_EPB * 48; t += E_TPB) {
    int m = t / 48, j = t - m * 48, o = j / 3, c = j - o * 3;
    float a = 0.f;
    for (int h = 0; h < 16; ++h) a = fmaf(VH[m*99 + h*3 + c], wv2[h*16 + o], a);
    VB[t] = a * GB[m * 16 + o];
  }
  __syncthreads();

  // ---- scatter-add aggregation at dst ----
#pragma unroll
  for (int t = 0; t < 8; ++t) {
    int col = t * 16 + m16;
#pragma unroll
    for (int i = 0; i < 8; ++i) {
      int mr = wid * 16 + i + 8 * hi;
      if (eblk + mr < nE)
        atomicAdd(agg_s + (size_t)Ldst[mr] * 128 + col, acc[t][i]);
    }
  }
  for (int t = tid; t < E_EPB * 48; t += E_TPB) {
    int m = t / 48, j = t - m * 48;
    if (eblk + m < nE) atomicAdd(agg_v + (size_t)Ldst[m] * 48 + j, VB[t]);
  }
  if (tid < E_EPB && (eblk + tid) < nE) atomicAdd(cnt + Ldst[tid], 1.0f);
}

// --------------------------- node (update+FF) kernel -----------------------
// 2 waves / block, 16 nodes / block; both waves split GEMM N-tiles.
__global__ void __launch_bounds__(64) gvp_node_kernel(
    const float* __restrict__ s_in, const float* __restrict__ v_in,
    const float* __restrict__ agg_s, const float* __restrict__ agg_v,
    const float* __restrict__ cnt,
    const float* __restrict__ whf0, const float* __restrict__ fsb0,
    const float* __restrict__ wvf0, const float* __restrict__ fgb0,
    const float* __restrict__ whf1, const float* __restrict__ fsb1,
    const float* __restrict__ wvf1, const float* __restrict__ fgb1,
    const float* __restrict__ Pf0, const float* __restrict__ Pg0,
    const float* __restrict__ Pf1, const float* __restrict__ Pg1,
    const float* __restrict__ n0s, const float* __restrict__ n0b,
    const float* __restrict__ n1s, const float* __restrict__ n1b,
    float* __restrict__ out_s, float* __restrict__ out_v, int nN) {
  __shared__ __align__(16) char smem[60672];
  float* SRES = (float*)(smem);              // [16][128] LN0(s) (residual base)
  float* VRES = (float*)(smem + 8192);       // [16][48]  LN0(v)
  float* PRE0 = (float*)(smem + 11264);      // [16][512] pre0; later pre1 [16][128]
  float* VH0  = (float*)(smem + 44032);      // [16][96] vh0; later vh1
  float* VN0  = (float*)(smem + 50176);      // [16][32] vn0; later vn1
  float* V1   = (float*)(smem + 52224);      // [16][96] v1; later v2 [16][48]
  float* GB   = (float*)(smem + 58368);      // [16][32] gates
  float* STAT = (float*)(smem + 60416);      // icnt[16] mu[16] rstd[16] vden[16]

  const int tid = threadIdx.x, lane = tid & 31, wid = tid >> 5;
  const int m16 = lane & 15, hi = lane >> 4;
  const int nblk = blockIdx.x * 16;

  if (tid < 16) {
    int nid = min(nblk + tid, nN - 1);
    STAT[tid] = 1.0f / fmaxf(cnt[nid], 1.0f);
  }
  __syncthreads();
  // ---- s: residual of mean-agg then LayerNorm0 ----
  for (int t = tid; t < 16 * 128; t += 64) {
    int m = t >> 7, j = t & 127;
    int nid = min(nblk + m, nN - 1);
    SRES[t] = s_in[(size_t)nid * 128 + j] + agg_s[(size_t)nid * 128 + j] * STAT[m];
  }
  __syncthreads();
  if (tid < 16) {
    float su = 0.f, sq = 0.f;
    for (int j = 0; j < 128; ++j) { float x = SRES[tid*128 + j]; su += x; sq += x*x; }
    float mu = su * (1.f/128.f);
    STAT[16 + tid] = mu;
    STAT[32 + tid] = rsqrtf(fmaxf(sq * (1.f/128.f) - mu*mu, 0.f) + 1e-5f);
  }
  __syncthreads();
  for (int t = tid; t < 16 * 128; t += 64) {
    int m = t >> 7, j = t & 127;
    SRES[t] = (SRES[t] - STAT[16 + m]) * STAT[32 + m] * n0s[j] + n0b[j];
  }
  // ---- v: residual then vector norm ----
  for (int t = tid; t < 16 * 48; t += 64) {
    int m = t / 48, j = t - m * 48;
    int nid = min(nblk + m, nN - 1);
    VRES[t] = v_in[(size_t)nid * 48 + j] + agg_v[(size_t)nid * 48 + j] * STAT[m];
  }
  __syncthreads();
  if (tid < 16) {
    float su = 0.f;
    for (int q = 0; q < 16; ++q) {
      float a = VRES[tid*48 + q*3], b = VRES[tid*48 + q*3+1], c = VRES[tid*48 + q*3+2];
      su += fmaxf(a*a + b*b + c*c, 1e-8f);
    }
    STAT[48 + tid] = rsqrtf(su * (1.f/16.f));
  }
  __syncthreads();
  for (int t = tid; t < 16 * 48; t += 64) VRES[t] *= STAT[48 + (t / 48)];
  __syncthreads();
  // ---- FF0: vh0 = v @ whf0 (16->32), vn0 ----
  for (int t = tid; t < 16 * 96; t += 64) {
    int m = t / 96, j = t - m * 96, h = j / 3, c = j - h * 3;
    float a = 0.f;
    for (int q = 0; q < 16; ++q) a = fmaf(VRES[m*48 + q*3 + c], whf0[q*32 + h], a);
    VH0[t] = a;
  }
  __syncthreads();
  for (int t = tid; t < 16 * 32; t += 64) {
    int m = t >> 5, h = t & 31;
    float a = VH0[m*96 + h*3], b = VH0[m*96 + h*3+1], c = VH0[m*96 + h*3+2];
    VN0[t] = sqrtf(fmaxf(a*a + b*b + c*c, 1e-8f));
  }
  __syncthreads();
  // ---- FF0 scalar GEMM [s|vn0](160) @ ws -> 512, each wave 16 tiles (2x8) --
  v8f acc[8];
  auto afS  = [&](int k) -> v2f { return ld2(SRES + m16 * 128 + k); };
  auto afV0 = [&](int k) -> v2f { return ld2(VN0 + m16 * 32 + (k - 128)); };
  for (int cb = 0; cb < 2; ++cb) {
#pragma unroll
    for (int t = 0; t < 8; ++t) acc[t] = zero8();
    wmma_gemm<8>(afS,  Pf0, 0,   128, 512, wid * 16 + cb * 8, acc);
    wmma_gemm<8>(afV0, Pf0, 128, 160, 512, wid * 16 + cb * 8, acc);
#pragma unroll
    for (int t = 0; t < 8; ++t) {
      int col = (wid * 16 + cb * 8 + t) * 16 + m16; float bb = fsb0[col];
#pragma unroll
      for (int i = 0; i < 8; ++i) PRE0[(i + 8*hi) * 512 + col] = acc[t][i] + bb;
    }
  }
  __syncthreads();
  // ---- gate0 = sigmoid(pre0) @ wsv (512->32), wave wid does tile wid ----
  v8f g = zero8();
  auto afsg = [&](int k) -> v2f {
    v2f x = ld2(PRE0 + m16 * 512 + k);
    v2f r; r[0] = sigf(x[0]); r[1] = sigf(x[1]); return r;
  };
  wmma_gemm<1>(afsg, Pg0, 0, 512, 32, wid, &g);
  {
    int col = wid * 16 + m16; float bb = fgb0[col];
#pragma unroll
    for (int i = 0; i < 8; ++i) GB[(i + 8*hi) * 32 + col] = sigf(g[i] + bb);
  }
  __syncthreads();
  // ---- v1 = (vh0 @ wvf0) * gate0 ----
  for (int t = tid; t < 16 * 96; t += 64) {
    int m = t / 96, j = t - m * 96, o = j / 3, c = j - o * 3;
    float a = 0.f;
    for (int h = 0; h < 32; ++h) a = fmaf(VH0[m*96 + h*3 + c], wvf0[h*32 + o], a);
    V1[t] = a * GB[m * 32 + o];
  }
  __syncthreads();
  // ---- FF1: vh1 = v1 @ whf1 (-> VH0 region), vn1 (-> VN0 region) ----
  for (int t = tid; t < 16 * 96; t += 64) {
    int m = t / 96, j = t - m * 96, h = j / 3, c = j - h * 3;
    float a = 0.f;
    for (int o = 0; o < 32; ++o) a = fmaf(V1[m*96 + o*3 + c], whf1[o*32 + h], a);
    VH0[t] = a;
  }
  __syncthreads();
  for (int t = tid; t < 16 * 32; t += 64) {
    int m = t >> 5, h = t & 31;
    float a = VH0[m*96 + h*3], b = VH0[m*96 + h*3+1], c = VH0[m*96 + h*3+2];
    VN0[t] = sqrtf(fmaxf(a*a + b*b + c*c, 1e-8f));
  }
  __syncthreads();
  // ---- FF1 scalar GEMM [relu(pre0)|vn1](544) -> 128, wave wid: 4 tiles ----
  v8f acc4[4];
#pragma unroll
  for (int t = 0; t < 4; ++t) acc4[t] = zero8();
  {
    auto afR = [&](int k) -> v2f {
      v2f x = ld2(PRE0 + m16 * 512 + k);
      v2f r; r[0] = fmaxf(x[0], 0.f); r[1] = fmaxf(x[1], 0.f); return r;
    };
    auto afV1 = [&](int k) -> v2f { return ld2(VN0 + m16 * 32 + (k - 512)); };
    wmma_gemm<4>(afR,  Pf1, 0,   512, 128, wid * 4, acc4);
    wmma_gemm<4>(afV1, Pf1, 512, 544, 128, wid * 4, acc4);
  }
  __syncthreads();
#pragma unroll
  for (int t = 0; t < 4; ++t) {
    int col = (wid * 4 + t) * 16 + m16; float bb = fsb1[col];
#pragma unroll
    for (int i = 0; i < 8; ++i) PRE0[(i + 8*hi) * 128 + col] = acc4[t][i] + bb;
  }
  __syncthreads();
  // ---- gate1 (acts=False: identity gate_in) ----
  g = zero8();
  auto afid = [&](int k) -> v2f { return ld2(PRE0 + m16 * 128 + k); };
  wmma_gemm<1>(afid, Pg1, 0, 128, 16, 0, &g);
  if (wid == 0) {
    float bb = fgb1[m16];
#pragma unroll
    for (int i = 0; i < 8; ++i) GB[(i + 8*hi) * 16 + m16] = sigf(g[i] + bb);
  }
  __syncthreads();
  // ---- v2 = (vh1 @ wvf1) * gate1  (-> V1 region, [16][48]) ----
  for (int t = tid; t < 16 * 48; t += 64) {
    int m = t / 48, j = t - m * 48, o = j / 3, c = j - o * 3;
    float a = 0.f;
    for (int h = 0; h < 32; ++h) a = fmaf(VH0[m*96 + h*3 + c], wvf1[h*16 + o], a);
    V1[m * 48 + j] = a * GB[m * 16 + o];
  }
  __syncthreads();
  // ---- residual + LayerNorm1 -> output ----
  if (tid < 16) {
    float su = 0.f, sq = 0.f;
    for (int j = 0; j < 128; ++j) {
      float x = SRES[tid*128 + j] + PRE0[tid*128 + j];
      su += x; sq += x * x;
    }
    float mu = su * (1.f/128.f);
    STAT[16 + tid] = mu;
    STAT[32 + tid] = rsqrtf(fmaxf(sq * (1.f/128.f) - mu*mu, 0.f) + 1e-5f);
    float sv = 0.f;
    for (int q = 0; q < 16; ++q) {
      float a = VRES[tid*48 + q*3]     + V1[tid*48 + q*3];
      float b = VRES[tid*48 + q*3 + 1] + V1[tid*48 + q*3 + 1];
      float c = VRES[tid*48 + q*3 + 2] + V1[tid*48 + q*3 + 2];
      sv += fmaxf(a*a + b*b + c*c, 1e-8f);
    }
    STAT[48 + tid] = rsqrtf(sv * (1.f/16.f));
  }
  __syncthreads();
  for (int t = tid; t < 16 * 128; t += 64) {
    int m = t >> 7, j = t & 127;
    int nid = nblk + m;
    if (nid < nN) {
      float x = SRES[t] + PRE0[t];
      out_s[(size_t)nid * 128 + j] = (x - STAT[16 + m]) * STAT[32 + m] * n1s[j] + n1b[j];
    }
  }
  for (int t = tid; t < 16 * 48; t += 64) {
    int m = t / 48, j = t - m * 48;
    int nid = nblk + m;
    if (nid < nN) out_v[(size_t)nid * 48 + j] = (VRES[t] + V1[t]) * STAT[48 + m];
  }
}

// --------------------------- host launcher ---------------------------------
extern "C" void kernel_launch(void* const* d_in, const int* in_sizes, int n_in,
                              void* d_out, int out_size, void* d_ws, size_t ws_size,
                              hipStream_t stream) {
  (void)n_in; (void)out_size; (void)ws_size;
  const float* s_in = (const float*)d_in[0];
  const float* v_in = (const float*)d_in[1];
  const int*   eidx = (const int*)d_in[2];
  const float* e_s  = (const float*)d_in[3];
  const float* e_v  = (const float*)d_in[4];
  // params in setup_inputs() insertion order: msg[0..2], ff[0..1] each
  // {wh, ws_w, ws_b, wv, wsv_w, wsv_b}, then norm0{scale,bias}, norm1{scale,bias}
  const float* P[40];
  for (int i = 5; i < 39; ++i) P[i] = (const float*)d_in[i];
  const float *m0_wh = P[5],  *m0_wsw = P[6],  *m0_wsb = P[7],  *m0_wv = P[8],  *m0_gw = P[9],  *m0_gb = P[10];
  const float *m1_wh = P[11], *m1_wsw = P[12], *m1_wsb = P[13], *m1_wv = P[14], *m1_gw = P[15], *m1_gb = P[16];
  const float *m2_wh = P[17], *m2_wsw = P[18], *m2_wsb = P[19], *m2_wv = P[20], *m2_gw = P[21], *m2_gb = P[22];
  const float *f0_wh = P[23], *f0_wsw = P[24], *f0_wsb = P[25], *f0_wv = P[26], *f0_gw = P[27], *f0_gb = P[28];
  const float *f1_wh = P[29], *f1_wsw = P[30], *f1_wsb = P[31], *f1_wv = P[32], *f1_gw = P[33], *f1_gb = P[34];
  const float *n0s = P[35], *n0b = P[36], *n1s = P[37], *n1b = P[38];

  const int nN = in_sizes[0] / 128;
  const int nE = in_sizes[2] / 2;

  float* ws = (float*)d_ws;
  size_t off = 0;
  float* agg_s = ws + off; off += (size_t)nN * 128;
  float* agg_v = ws + off; off += (size_t)nN * 48;
  float* cntb  = ws + off; off += (size_t)nN;
  off = (off + 31) & ~(size_t)31;
  float* Pws0 = ws + off; off += 324 * 128;
  float* Pws1 = ws + off; off += 144 * 128;
  float* Pws2 = ws + off; off += 144 * 128;
  float* Psv0 = ws + off; off += 128 * 16;
  float* Psv1 = ws + off; off += 128 * 16;
  float* Psv2 = ws + off; off += 128 * 16;
  float* Pf0  = ws + off; off += 160 * 512;
  float* Pg0  = ws + off; off += 512 * 32;
  float* Pf1  = ws + off; off += 544 * 128;
  float* Pg1  = ws + off; off += 128 * 16;

  float* out_s = (float*)d_out;
  float* out_v = out_s + (size_t)nN * 128;

  // 1) zero aggregation buffers (contiguous at ws base)
  long zn = (long)nN * (128 + 48 + 1);
  zero_kernel<<<(int)((zn + 255) / 256), 256, 0, stream>>>(ws, zn);

  // 2) pack GEMM weights into k-pair interleaved layout
  auto packs = [&](const float* W, float* Bp, int K, int Kp4, int N) {
    int total = Kp4 * N;
    pack_kernel<<<(total + 255) / 256, 256, 0, stream>>>(W, Bp, K, Kp4, N);
  };
  packs(m0_wsw, Pws0, 321, 324, 128);
  packs(m1_wsw, Pws1, 144, 144, 128);
  packs(m2_wsw, Pws2, 144, 144, 128);
  packs(m0_gw,  Psv0, 128, 128, 16);
  packs(m1_gw,  Psv1, 128, 128, 16);
  packs(m2_gw,  Psv2, 128, 128, 16);
  packs(f0_wsw, Pf0,  160, 160, 512);
  packs(f0_gw,  Pg0,  512, 512, 32);
  packs(f1_wsw, Pf1,  544, 544, 128);
  packs(f1_gw,  Pg1,  128, 128, 16);

  // 3) edge message + scatter-add
  gvp_edge_kernel<<<(nE + E_EPB - 1) / E_EPB, E_TPB, 0, stream>>>(
      s_in, v_in, eidx, e_s, e_v,
      m0_wh, m0_wsb, m0_wv, m0_gb,
      m1_wh, m1_wsb, m1_wv, m1_gb,
      m2_wh, m2_wsb, m2_wv, m2_gb,
      Pws0, Pws1, Pws2, Psv0, Psv1, Psv2,
      agg_s, agg_v, cntb, nE);

  // 4) node update: mean-agg + LN + feedforward + LN
  gvp_node_kernel<<<(nN + 15) / 16, 64, 0, stream>>>(
      s_in, v_in, agg_s, agg_v, cntb,
      f0_wh, f0_wsb, f0_wv, f0_gb,
      f1_wh, f1_wsb, f1_wv, f1_gb,
      Pf0, Pg0, Pf1, Pg1,
      n0s, n0b, n1s, n1b,
      out_s, out_v, nN);
}